// TranslationsIdenticalCost_62027917688869
// MI455X (gfx1250) — compile-verified
//
#include <hip/hip_runtime.h>
#include <hip/hip_bf16.h>

typedef __attribute__((ext_vector_type(2))) float v2f;
typedef __attribute__((ext_vector_type(8))) float v8f;

#define MPTS   4096     // number of control points (M)
#define CHUNK  2048     // control points staged in LDS per pass (64KB)
#define WAVES  8        // waves per block (wave32)
#define BLOCK  (WAVES * 32)

// exp(-sq/sigma^2) with sigma=0.5 -> exp(8*D) where D = -0.5*sq
// exp(8*D) = exp2(D * 8 * log2(e))
#define K8 11.5415602f  // 8 * 1.4426950408889634

__global__ __launch_bounds__(BLOCK)
void rbf_conv_kernel(const float* __restrict__ GD,
                     const float* __restrict__ Cont,
                     const float* __restrict__ Pts,
                     float* __restrict__ Out)
{
    // LDS staging: fused control-point records
    __shared__ float4 sY[CHUNK];   // {yx, yy, yz, -0.5*||y||^2}
    __shared__ float4 sC[CHUNK];   // {cx, cy, cz, 0}

    const int lane = threadIdx.x & 31;
    const int wave = threadIdx.x >> 5;
    const bool hi  = lane >= 16;        // lanes 16-31 hold K=2/K=3 slots of A/B
    const int ln16 = lane & 15;

    // This wave owns a tile of 16 points.
    const int i0 = (blockIdx.x * WAVES + wave) * 16;

    // Every lane loads point (i0 + lane%16) fully.
    const int p = i0 + ln16;
    const float px = Pts[p * 3 + 0];
    const float py = Pts[p * 3 + 1];
    const float pz = Pts[p * 3 + 2];
    const float x2 = px * px + py * py + pz * pz;

    // A matrix (16x4 f32, persistent): rows [x, y, z, 1]
    //   vgpr0: K=0 (low lanes) = x   | K=2 (high lanes) = z
    //   vgpr1: K=1 (low lanes) = y   | K=3 (high lanes) = 1
    v2f A;
    A.x = hi ? pz   : px;
    A.y = hi ? 1.0f : py;

    // Build cinit[i][j] = -0.5*||x_i||^2 (broadcast over j) with one WMMA:
    //   A2[i][0] = -0.5*x2_i, rest 0;  B2[0][j] = 1, rest 0.
    v2f A2, B2;
    A2.x = hi ? 0.0f : -0.5f * x2;
    A2.y = 0.0f;
    B2.x = hi ? 0.0f : 1.0f;
    B2.y = 0.0f;
    v8f zero = {};
    v8f cinit = __builtin_amdgcn_wmma_f32_16x16x4_f32(
        false, A2, false, B2, (short)0, zero, false, false);

    // 24 accumulators: acc{0,1,2}[r] = sum_j K(i_r, j) * c_j.{x,y,z}
    v8f acc0 = {}, acc1 = {}, acc2 = {};

    for (int base = 0; base < MPTS; base += CHUNK) {
        // Cooperative fill of LDS with fused records (no divergence: 2048 % 256 == 0)
        for (int j = threadIdx.x; j < CHUNK; j += BLOCK) {
            const int g = base + j;
            const float yx = GD[g * 3 + 0];
            const float yy = GD[g * 3 + 1];
            const float yz = GD[g * 3 + 2];
            sY[j] = make_float4(yx, yy, yz, -0.5f * (yx * yx + yy * yy + yz * yz));
            sC[j] = make_float4(Cont[g * 3 + 0], Cont[g * 3 + 1], Cont[g * 3 + 2], 0.0f);
        }
        __syncthreads();

        // Sweep 16-wide j-tiles of this chunk
        #pragma unroll 4
        for (int t = 0; t < CHUNK; t += 16) {
            const int jj = t + ln16;
            const float4 q  = sY[jj];   // ds_load_b128
            const float4 cc = sC[jj];   // ds_load_b128

            // B matrix (4x16 f32): columns [yx, yy, yz, -0.5*y2]
            v2f B;
            B.x = hi ? q.z : q.x;
            B.y = hi ? q.w : q.y;

            // D[i][j] = x.y - 0.5*y2 - 0.5*x2 = -0.5 * ||x - y||^2
            v8f D = __builtin_amdgcn_wmma_f32_16x16x4_f32(
                false, A, false, B, (short)0, cinit, false, false);

            #pragma unroll
            for (int r = 0; r < 8; ++r) {
                const float e = __builtin_amdgcn_exp2f(D[r] * K8); // v_exp_f32
                acc0[r] = fmaf(e, cc.x, acc0[r]);
                acc1[r] = fmaf(e, cc.y, acc1[r]);
                acc2[r] = fmaf(e, cc.z, acc2[r]);
            }
        }
        __syncthreads();
    }

    // Reduce over the 16 lanes of each half (each lane held one j-class).
    // xor masks 1,2,4,8 never cross bit 4, so halves stay independent.
    #pragma unroll
    for (int off = 8; off >= 1; off >>= 1) {
        #pragma unroll
        for (int r = 0; r < 8; ++r) {
            acc0[r] += __shfl_xor(acc0[r], off, 32);
            acc1[r] += __shfl_xor(acc1[r], off, 32);
            acc2[r] += __shfl_xor(acc2[r], off, 32);
        }
    }

    // Lane 0 writes points i0..i0+7 (M=r), lane 16 writes i0+8..i0+15 (M=8+r).
    if (ln16 == 0) {
        const int ibase = i0 + (hi ? 8 : 0);
        #pragma unroll
        for (int r = 0; r < 8; ++r) {
            Out[(ibase + r) * 3 + 0] = acc0[r];
            Out[(ibase + r) * 3 + 1] = acc1[r];
            Out[(ibase + r) * 3 + 2] = acc2[r];
        }
    }
}

extern "C" void kernel_launch(void* const* d_in, const int* in_sizes, int n_in,
                              void* d_out, int out_size, void* d_ws, size_t ws_size,
                              hipStream_t stream) {
    const float* GD   = (const float*)d_in[0];   // [4096*3]
    const float* Cont = (const float*)d_in[1];   // [4096*3]
    const float* Pts  = (const float*)d_in[2];   // [65536, 3]
    float* Out        = (float*)d_out;           // [65536, 3]

    const int N = in_sizes[2] / 3;               // 65536
    const int blocks = N / (16 * WAVES);         // 512
    rbf_conv_kernel<<<blocks, BLOCK, 0, stream>>>(GD, Cont, Pts, Out);
}